// GraphAugmentation_9165460210004
// MI455X (gfx1250) — compile-verified
//
#include <hip/hip_runtime.h>

typedef float v2f __attribute__((ext_vector_type(2)));
typedef float v8f __attribute__((ext_vector_type(8)));

namespace {
constexpr int kH = 512, kW = 512, kC = 16;
constexpr int kHW = kH * kW;
constexpr int kTilesPerRow = kW / 16;            // 32
constexpr int kTiles = 4 * kH * kTilesPerRow;    // 65536 tiles of 16 pixels
constexpr int kBlocks = 2048, kThreads = 256, kWavesPerBlk = kThreads / 32;
constexpr int kTPW = kTiles / (kBlocks * kWavesPerBlk);  // 4 tiles per wave

// OFFSETS (dy,dx); rolled[h][w] = M[(h-dy)%H][(w-dx)%W]
__device__ __constant__ int c_dy[8] = {-4, -4, -4, -3, -2, 2, 3, 4};
__device__ __constant__ int c_dx[8] = {-4, -1, 2, 4, -3, 3, -2, 4};

__device__ __forceinline__ v8f wmma4_f32(v2f a, v2f b, v8f c) {
  // D = A(16x4,f32) * B(4x16,f32) + C(16x16,f32)
  return __builtin_amdgcn_wmma_f32_16x16x4_f32(false, a, false, b, (short)0, c,
                                               false, false);
}

__device__ __forceinline__ void tile_coords(int t, int& b, int& h, int& w0) {
  b = t >> 14;               // / (512*32)
  int rem = t & 16383;
  h = rem >> 5;              // / 32
  w0 = (rem & 31) << 4;      // *16
}
}  // namespace

// -------- Pass 1: M = mw @ x + mb, staged to workspace --------
__global__ __launch_bounds__(256) void ga_pass1_mconv(
    const float* __restrict__ x, const float* __restrict__ mw,
    const float* __restrict__ mb, float* __restrict__ M) {
  const int lane = threadIdx.x & 31;
  const int p = lane & 15;        // pixel / matrix row within tile
  const int half = lane >> 4;     // lane-half selects K pair / +8 on D rows
  const int kofs = half * 2;

  // A fragments: mw[16x16], 4 chunks of K=4
  v2f wA[4];
#pragma unroll
  for (int kk = 0; kk < 4; ++kk) {
    wA[kk].x = mw[p * 16 + kk * 4 + kofs];
    wA[kk].y = mw[p * 16 + kk * 4 + kofs + 1];
  }
  float bias[8];
#pragma unroll
  for (int r = 0; r < 8; ++r) bias[r] = mb[r + 8 * half];

  const int numWaves = kBlocks * kWavesPerBlk;
  const int wgid = blockIdx.x * kWavesPerBlk + (threadIdx.x >> 5);
#pragma unroll 1
  for (int k = 0; k < kTPW; ++k) {
    int b, h, w0;
    tile_coords(wgid + k * numWaves, b, h, w0);
    const float* xb = x + (size_t)b * kC * kHW + h * kW + w0 + p;

    v2f vB[4];
#pragma unroll
    for (int kk = 0; kk < 4; ++kk) {
      int c0 = kk * 4 + kofs;
      vB[kk].x = xb[(size_t)c0 * kHW];
      vB[kk].y = xb[(size_t)(c0 + 1) * kHW];
    }
    v8f acc;
#pragma unroll
    for (int r = 0; r < 8; ++r) acc[r] = bias[r];
#pragma unroll
    for (int kk = 0; kk < 4; ++kk) acc = wmma4_f32(wA[kk], vB[kk], acc);

    float* Mb = M + (size_t)b * kC * kHW + h * kW + w0 + p;
#pragma unroll
    for (int r = 0; r < 8; ++r) Mb[(size_t)(r + 8 * half) * kHW] = acc[r];
  }
}

// -------- Pass 2: agg = mean of 8 rolled M; gated fusion --------
__global__ __launch_bounds__(256) void ga_pass2_gate(
    const float* __restrict__ x, const float* __restrict__ M,
    const float* __restrict__ g1w, const float* __restrict__ g1b,
    const float* __restrict__ g2w, const float* __restrict__ g2b,
    float* __restrict__ out) {
  __shared__ float s_agg[kWavesPerBlk][16][16];     // per-wave private slices
  __shared__ float s_hidden[kWavesPerBlk][32][16];

  const int lane = threadIdx.x & 31;
  const int wave = threadIdx.x >> 5;
  const int p = lane & 15;
  const int half = lane >> 4;
  const int kofs = half * 2;

  // A fragments: g1w[32x32] -> two 16-row M-tiles, 8 K-chunks each
  v2f g1A[2][8];
#pragma unroll
  for (int t = 0; t < 2; ++t)
#pragma unroll
    for (int kk = 0; kk < 8; ++kk) {
      g1A[t][kk].x = g1w[(16 * t + p) * 32 + kk * 4 + kofs];
      g1A[t][kk].y = g1w[(16 * t + p) * 32 + kk * 4 + kofs + 1];
    }
  // A fragments: g2w[16x32], 8 K-chunks
  v2f g2A[8];
#pragma unroll
  for (int kk = 0; kk < 8; ++kk) {
    g2A[kk].x = g2w[p * 32 + kk * 4 + kofs];
    g2A[kk].y = g2w[p * 32 + kk * 4 + kofs + 1];
  }
  float b1[2][8], b2[8];
#pragma unroll
  for (int t = 0; t < 2; ++t)
#pragma unroll
    for (int r = 0; r < 8; ++r) b1[t][r] = g1b[16 * t + r + 8 * half];
#pragma unroll
  for (int r = 0; r < 8; ++r) b2[r] = g2b[r + 8 * half];

  const int numWaves = kBlocks * kWavesPerBlk;
  const int wgid = blockIdx.x * kWavesPerBlk + wave;
#pragma unroll 1
  for (int k = 0; k < kTPW; ++k) {
    int b, h, w0;
    tile_coords(wgid + k * numWaves, b, h, w0);

    // Per-lane gather bases for the 8 circular shifts.
    int base[8];
#pragma unroll
    for (int i = 0; i < 8; ++i) {
      int hs = (h - c_dy[i] + kH) & (kH - 1);
      int ws = (w0 + p - c_dx[i] + kW) & (kW - 1);
      base[i] = hs * kW + ws;
    }

    // B fragments for concat[32 ch][16 px]: chunks 0..3 = x, 4..7 = agg.
    v2f vB[8];
    const float* xb = x + (size_t)b * kC * kHW + h * kW + w0 + p;
    const float* Mb = M + (size_t)b * kC * kHW;
#pragma unroll
    for (int q = 0; q < 4; ++q) {
#pragma unroll
      for (int j = 0; j < 2; ++j) {
        int c = q * 4 + kofs + j;                 // channel this lane owns
        float xv = xb[(size_t)c * kHW];           // x channel c
        const float* Mc = Mb + (size_t)c * kHW;   // agg channel c
        float s = 0.f;
#pragma unroll
        for (int i = 0; i < 8; ++i) s += Mc[base[i]];
        s *= 0.125f;                              // softmax of equal affinities
        if (j == 0) { vB[q].x = xv; vB[4 + q].x = s; }
        else        { vB[q].y = xv; vB[4 + q].y = s; }
        s_agg[wave][c][p] = s;                    // stash for D-layout readback
      }
    }

    // hidden = relu(g1w @ concat + g1b): two 16-row D tiles, K=32.
#pragma unroll
    for (int t = 0; t < 2; ++t) {
      v8f acc;
#pragma unroll
      for (int r = 0; r < 8; ++r) acc[r] = b1[t][r];
#pragma unroll
      for (int kk = 0; kk < 8; ++kk) acc = wmma4_f32(g1A[t][kk], vB[kk], acc);
#pragma unroll
      for (int r = 0; r < 8; ++r)
        s_hidden[wave][16 * t + r + 8 * half][p] = fmaxf(acc[r], 0.f);
    }

    // Re-read hidden as B fragments (D->B transpose through LDS).
    v2f vH[8];
#pragma unroll
    for (int kk = 0; kk < 8; ++kk) {
      vH[kk].x = s_hidden[wave][kk * 4 + kofs][p];
      vH[kk].y = s_hidden[wave][kk * 4 + kofs + 1][p];
    }

    // gate = sigmoid(g2w @ hidden + g2b), K=32.
    v8f acc;
#pragma unroll
    for (int r = 0; r < 8; ++r) acc[r] = b2[r];
#pragma unroll
    for (int kk = 0; kk < 8; ++kk) acc = wmma4_f32(g2A[kk], vH[kk], acc);

    // out = agg * gate  (agg re-read in D layout).
    float* ob = out + (size_t)b * kC * kHW + h * kW + w0 + p;
#pragma unroll
    for (int r = 0; r < 8; ++r) {
      float g = 1.0f / (1.0f + __expf(-acc[r]));
      int m = r + 8 * half;
      ob[(size_t)m * kHW] = s_agg[wave][m][p] * g;
    }
  }
}

extern "C" void kernel_launch(void* const* d_in, const int* in_sizes, int n_in,
                              void* d_out, int out_size, void* d_ws,
                              size_t ws_size, hipStream_t stream) {
  // setup_inputs order: x, qw, qb, kw, kb, mw, mb, scaling, g1w, g1b, g2w, g2b
  // (qw/qb/kw/kb/scaling are mathematically dead: softmax of equal affinities
  //  is exactly uniform 1/8.)
  const float* x   = (const float*)d_in[0];
  const float* mw  = (const float*)d_in[5];
  const float* mb  = (const float*)d_in[6];
  const float* g1w = (const float*)d_in[8];
  const float* g1b = (const float*)d_in[9];
  const float* g2w = (const float*)d_in[10];
  const float* g2b = (const float*)d_in[11];
  float* out = (float*)d_out;
  float* M = (float*)d_ws;  // 4*16*512*512 floats = 64 MB staging for M

  ga_pass1_mconv<<<kBlocks, kThreads, 0, stream>>>(x, mw, mb, M);
  ga_pass2_gate<<<kBlocks, kThreads, 0, stream>>>(x, M, g1w, g1b, g2w, g2b,
                                                  out);
}